// CapsuleLayer_65146063946632
// MI455X (gfx1250) — compile-verified
//
#include <hip/hip_runtime.h>
#include <math.h>

#define BATCH    64
#define N_CAPS   32
#define N_ROUTES 2048
#define IN_DIM   16
#define OUT_DIM  32
#define NT       16                 // routes per workgroup
#define XSTRIDE  (NT * IN_DIM + 4)  // padded x-chunk stride (dwords) to spread LDS banks

typedef __attribute__((ext_vector_type(2))) float v2f;
typedef __attribute__((ext_vector_type(8))) float v8f;

// Exact pointee type expected by the async-LDS builtin (per hipcc diagnostic):
// 'int __attribute__((vector_size(16)))' in AS1 (global) / AS3 (LDS).
typedef int v4i_t __attribute__((__vector_size__(16)));
typedef __attribute__((address_space(1))) v4i_t* gv4i_p;
typedef __attribute__((address_space(3))) v4i_t* lv4i_p;

#if defined(__has_builtin)
#if __has_builtin(__builtin_amdgcn_global_load_async_to_lds_b128)
#define HAVE_ASYNC_LDS 1
#endif
#if __has_builtin(__builtin_amdgcn_s_wait_asynccnt)
#define HAVE_ASYNCCNT_WAIT 1
#endif
#endif

// 16-byte global -> LDS staging copy; async (ASYNCcnt-tracked) when available.
__device__ __forceinline__ void stage16(const float* __restrict__ g, float* l)
{
#ifdef HAVE_ASYNC_LDS
    __builtin_amdgcn_global_load_async_to_lds_b128((gv4i_p)g, (lv4i_p)l, 0, 0);
#else
    *(float4*)l = *(const float4*)g;
#endif
}

__device__ __forceinline__ void stage_wait()
{
#ifdef HAVE_ASYNCCNT_WAIT
    __builtin_amdgcn_s_wait_asynccnt(0);
#endif
    __syncthreads();
}

// One fused routing pass:
//   a[n]   = priors[n,:] . out_prev          (agreement of previous iteration)
//   B[n]  += a[n]
//   e[n]   = exp(B[n])
//   Z     += e[n];  S[:] += e[n]*priors[n,:]
// priors are recomputed on the fly with V_WMMA_F32_16X16X4_F32 (never hit HBM).
__global__ __launch_bounds__(256)
void caps_accum(const float* __restrict__ x, const float* __restrict__ W,
                float* __restrict__ Bmat, const float* __restrict__ outPrev,
                float* __restrict__ S, float* __restrict__ Z, int iter)
{
    extern __shared__ float smem[];
    float* sp  = smem;                              // priors  [NT][BATCH][OUT_DIM]  128 KB
    float* xs  = sp + NT * BATCH * OUT_DIM;         // x stage [BATCH][XSTRIDE]      ~65 KB
    float* wsm = xs + BATCH * XSTRIDE;              // W stage [NT][IN_DIM][OUT_DIM]  32 KB

    const int c    = blockIdx.y;
    const int nb   = blockIdx.x;
    const int t    = threadIdx.x;
    const int wid  = t >> 5;
    const int lane = t & 31;
    const int laneRow = lane & 15;
    const int laneHi  = lane >> 4;          // 0 or 1
    const int m0 = (wid >> 1) << 4;         // batch tile: 0,16,32,48
    const int o0 = (wid & 1) << 4;          // out tile:   0,16
    const int rbase = nb * NT;

    // ---------------- Stage x & W blocks into LDS (coalesced, async) ----------------
    // W block: NT*16*32 floats, fully contiguous in global memory (32 KB).
    const float* wsrc = W + ((size_t)c * N_ROUTES + rbase) * (IN_DIM * OUT_DIM);
#pragma unroll
    for (int j = 0; j < (NT * IN_DIM * OUT_DIM / 4) / 256; ++j) {
        const int idx = t + j * 256;                    // float4 index
        stage16(wsrc + idx * 4, wsm + idx * 4);
    }
    // x block: per batch row a contiguous NT*16-float chunk (1 KB), 64 chunks.
#pragma unroll
    for (int j = 0; j < (BATCH * NT * IN_DIM / 4) / 256; ++j) {
        const int idx = t + j * 256;                    // float4 index
        const int b   = idx >> 6;                       // NT*IN_DIM/4 = 64 float4 per chunk
        const int pos = idx & 63;
        stage16(x + ((size_t)b * N_ROUTES + rbase) * IN_DIM + pos * 4,
                xs + b * XSTRIDE + pos * 4);
    }
    stage_wait();

    // ---------------- Phase 1: priors via fp32 WMMA ----------------
    // A (16x4): lane holds x[b = m0+laneRow][k0+2*laneHi .. +1]   (ds b64)
    // B (4x16): lane holds W[k0+2*laneHi..+1][o0+laneRow]         (2x ds b32)
    const float* xlane = xs + (m0 + laneRow) * XSTRIDE + 2 * laneHi;
    const float* wlane = wsm + (2 * laneHi) * OUT_DIM + (o0 + laneRow);
    for (int n = 0; n < NT; ++n) {
        const float* xp = xlane + n * IN_DIM;
        const float* wp = wlane + n * (IN_DIM * OUT_DIM);
        v8f acc = {};
#pragma unroll
        for (int k0 = 0; k0 < IN_DIM; k0 += 4) {
            v2f afrag = *(const v2f*)(xp + k0);
            v2f bfrag;
            bfrag.x = wp[(k0)     * OUT_DIM];
            bfrag.y = wp[(k0 + 1) * OUT_DIM];
            acc = __builtin_amdgcn_wmma_f32_16x16x4_f32(
                false, afrag, false, bfrag, (short)0, acc, false, false);
        }
        // C layout: VGPR r8 -> row m0 + r8 + 8*laneHi, col o0 + laneRow
        float* dst = sp + n * (BATCH * OUT_DIM) + (m0 + 8 * laneHi) * OUT_DIM + (o0 + laneRow);
#pragma unroll
        for (int r8 = 0; r8 < 8; ++r8)
            dst[r8 * OUT_DIM] = acc[r8];
    }
    __syncthreads();

    // ---------------- Phase 2: routing accumulation ----------------
    // 4 threads per batch row; each owns 8 of the 32 output dims.
    const int b  = t >> 2;
    const int q  = t & 3;
    const int ob = q << 3;
    const int bc = b * N_CAPS + c;

    float op[8];
#pragma unroll
    for (int j = 0; j < 8; ++j) op[j] = 0.0f;
    if (iter > 0) {
        const float4 p0 = *(const float4*)(outPrev + bc * OUT_DIM + ob);
        const float4 p1 = *(const float4*)(outPrev + bc * OUT_DIM + ob + 4);
        op[0]=p0.x; op[1]=p0.y; op[2]=p0.z; op[3]=p0.w;
        op[4]=p1.x; op[5]=p1.y; op[6]=p1.z; op[7]=p1.w;
    }

    float sacc[8];
#pragma unroll
    for (int j = 0; j < 8; ++j) sacc[j] = 0.0f;
    float zacc = 0.0f;

    float* Brow = Bmat + (size_t)bc * N_ROUTES + rbase;
    for (int n = 0; n < NT; ++n) {
        const float* pr = sp + n * (BATCH * OUT_DIM) + b * OUT_DIM + ob;
        const float4 p0 = *(const float4*)pr;
        const float4 p1 = *(const float4*)(pr + 4);
        float pv[8] = {p0.x, p0.y, p0.z, p0.w, p1.x, p1.y, p1.z, p1.w};

        float dot = 0.0f;
#pragma unroll
        for (int j = 0; j < 8; ++j) dot = fmaf(pv[j], op[j], dot);
        dot += __shfl_xor(dot, 1);     // reduce across the 4 q-lanes (wave32)
        dot += __shfl_xor(dot, 2);

        const float Bn = Brow[n] + dot;
        if (q == 0) Brow[n] = Bn;      // persist updated logit
        const float e = expf(Bn);      // unnormalized softmax weight
#pragma unroll
        for (int j = 0; j < 8; ++j) sacc[j] = fmaf(e, pv[j], sacc[j]);
        zacc += e;
    }

    float* Sp = S + bc * OUT_DIM + ob;
#pragma unroll
    for (int j = 0; j < 8; ++j) atomicAdd(Sp + j, sacc[j]);
    if (q == 0) atomicAdd(Z + bc, zacc);
}

// out = squash(S / Z); feeds next iteration and (on the last pass) d_out.
__global__ __launch_bounds__(256)
void caps_finalize(const float* __restrict__ S, const float* __restrict__ Z,
                   float* __restrict__ outPrev, float* __restrict__ out, int writeFinal)
{
    const int bc = blockIdx.x * blockDim.x + threadIdx.x;
    if (bc >= BATCH * N_CAPS) return;
    const float inv = 1.0f / Z[bc];
    float v[OUT_DIM];
    float sq = 0.0f;
#pragma unroll
    for (int j = 0; j < OUT_DIM; ++j) {
        v[j] = S[bc * OUT_DIM + j] * inv;
        sq = fmaf(v[j], v[j], sq);
    }
    const float scale = sq / ((1.0f + sq) * sqrtf(sq));
#pragma unroll
    for (int j = 0; j < OUT_DIM; ++j) {
        const float o = v[j] * scale;
        outPrev[bc * OUT_DIM + j] = o;
        if (writeFinal) out[bc * OUT_DIM + j] = o;
    }
}

extern "C" void kernel_launch(void* const* d_in, const int* in_sizes, int n_in,
                              void* d_out, int out_size, void* d_ws, size_t ws_size,
                              hipStream_t stream)
{
    const float* x = (const float*)d_in[0];          // [64, 2048, 16]
    const float* W = (const float*)d_in[1];          // [32, 2048, 16, 32]
    float* out = (float*)d_out;                      // [64, 32, 32]

    // Workspace layout (floats): B logits | S | Z | outPrev  (~16.9 MB total)
    float* B       = (float*)d_ws;                              // 64*32*2048
    float* S       = B + (size_t)BATCH * N_CAPS * N_ROUTES;     // 64*32*32
    float* Z       = S + (size_t)BATCH * N_CAPS * OUT_DIM;      // 64*32
    float* outPrev = Z + (size_t)BATCH * N_CAPS;                // 64*32*32

    (void)hipMemsetAsync(B, 0, sizeof(float) * (size_t)BATCH * N_CAPS * N_ROUTES, stream);

    const dim3 grid(N_ROUTES / NT, N_CAPS);
    const size_t shmem =
        ((size_t)NT * BATCH * OUT_DIM + (size_t)BATCH * XSTRIDE +
         (size_t)NT * IN_DIM * OUT_DIM) * sizeof(float);        // ~225 KB

    for (int it = 0; it < 3; ++it) {
        // S and Z are contiguous: one memset clears both accumulators.
        (void)hipMemsetAsync(S, 0,
                             sizeof(float) * ((size_t)BATCH * N_CAPS * OUT_DIM + BATCH * N_CAPS),
                             stream);
        caps_accum<<<grid, 256, shmem, stream>>>(x, W, B, outPrev, S, Z, it);
        caps_finalize<<<(BATCH * N_CAPS + 255) / 256, 256, 0, stream>>>(
            S, Z, outPrev, out, it == 2 ? 1 : 0);
    }
}